// MultiScaleRetention_41575283425483
// MI455X (gfx1250) — compile-verified
//
#include <hip/hip_runtime.h>
#include <math.h>

// MultiScaleRetention for MI455X (gfx1250, wave32).
//   K1 k_proj : P[h] = q @ Wqkv[h] + bqkv[h]          (f32 WMMA GEMM, N=192/head)
//   K2 k_scan : chunkwise retention recurrence per (b,h), fused GroupNorm stats
//   K3 k_gate : Z = swish(q@Wg+bg) * ((X-mu)*istd + beta)   (f32 WMMA GEMM)
//   K4 k_outp : out = Z @ Wo + bo                     (f32 WMMA GEMM)
// Workspace: P = 25,165,824 f32 (100.7MB), X = 8,388,608 f32 (33.5MB),
//            stats = 256 f32; Z aliases P (dead after K2). Needs ~134.3MB ws.

typedef float v2f __attribute__((ext_vector_type(2)));
typedef float v8f __attribute__((ext_vector_type(8)));

#define DIMS   512
#define HDIMS  64
#define NHEAD  8
#define BATCH  16
#define SEQ    1024
#define MROWS  (BATCH * SEQ)   // 16384
#define NCHUNK (SEQ / 2)       // 512
#define PROJ_N 192

#define AS_STRIDE 34           // A tile row stride (b64-aligned, bank-spread)
#define BS_STRIDE 160          // B k-pair row stride; 160 % 64 == 32 -> the
                               // upper half-wave (K+2) hits the other bank half

// ---------------- common 128x64 f32 WMMA tile GEMM core -----------------
// 256 threads = 8 waves; wave w owns rows w*16..w*16+15 and all 64 cols:
// one A fragment feeds four v_wmma_f32_16x16x4_f32 per K-step (A reuse x4).
// B is stored K-pair interleaved so each fragment is a single ds_load_b64.
// Global->LDS is software pipelined (next slab prefetched into registers).
__device__ __forceinline__ void gemm_block_128x64(
    const float* __restrict__ A, int lda,
    const float* __restrict__ B, int ldb,
    int row0, int col0, int K,
    float* As /*128 x AS_STRIDE*/, float* Bs /*16 x BS_STRIDE*/, v8f c[4])
{
  const int tid  = threadIdx.x;
  const int lane = tid & 31;
  const int wid  = tid >> 5;
  const int ml   = (wid << 4) + (lane & 15);
  const int nl2  = (lane & 15) << 1;
  // per-thread global-load coordinates
  const int ra = tid >> 3, ca = (tid & 7) << 2;   // A: 128x32, +32 rows per i
  const int rb = tid >> 4, cb = (tid & 15) << 2;  // B: 32x64,  +16 rows per i

  float4 ar[4], br[2];
  #pragma unroll
  for (int i = 0; i < 4; ++i)
    ar[i] = *(const float4*)(A + (size_t)(row0 + ra + (i << 5)) * lda + ca);
  #pragma unroll
  for (int i = 0; i < 2; ++i)
    br[i] = *(const float4*)(B + (size_t)(rb + (i << 4)) * ldb + (col0 + cb));

  for (int k0 = 0; k0 < K; k0 += 32) {
    __syncthreads();
    #pragma unroll
    for (int i = 0; i < 4; ++i) {
      float* d = As + (ra + (i << 5)) * AS_STRIDE + ca;
      d[0] = ar[i].x; d[1] = ar[i].y; d[2] = ar[i].z; d[3] = ar[i].w;
    }
    #pragma unroll
    for (int i = 0; i < 2; ++i) {
      const int r = rb + (i << 4);
      float* d = Bs + (r >> 1) * BS_STRIDE + (r & 1);
      d[(cb << 1)]     = br[i].x;
      d[(cb << 1) + 2] = br[i].y;
      d[(cb << 1) + 4] = br[i].z;
      d[(cb << 1) + 6] = br[i].w;
    }
    __syncthreads();
    if (k0 + 32 < K) {                      // prefetch next K-slab
      #pragma unroll
      for (int i = 0; i < 4; ++i)
        ar[i] = *(const float4*)(A + (size_t)(row0 + ra + (i << 5)) * lda
                                 + (k0 + 32 + ca));
      #pragma unroll
      for (int i = 0; i < 2; ++i)
        br[i] = *(const float4*)(B + (size_t)(k0 + 32 + rb + (i << 4)) * ldb
                                 + (col0 + cb));
    }
    #pragma unroll
    for (int kk = 0; kk < 32; kk += 4) {
      // ISA 16x4 f32 A layout: lanes 0-15 hold K={0,1}, lanes 16-31 K={2,3}
      const int kb = kk + ((lane >> 4) << 1);
      const v2f a = *(const v2f*)(As + ml * AS_STRIDE + kb);
      const float* bp = Bs + ((kk >> 1) + (lane >> 4)) * BS_STRIDE + nl2;
      const v2f b0 = *(const v2f*)(bp);
      const v2f b1 = *(const v2f*)(bp + 32);
      const v2f b2 = *(const v2f*)(bp + 64);
      const v2f b3 = *(const v2f*)(bp + 96);
      c[0] = __builtin_amdgcn_wmma_f32_16x16x4_f32(false, a, false, b0, (short)0, c[0], false, false);
      c[1] = __builtin_amdgcn_wmma_f32_16x16x4_f32(false, a, false, b1, (short)0, c[1], false, false);
      c[2] = __builtin_amdgcn_wmma_f32_16x16x4_f32(false, a, false, b2, (short)0, c[2], false, false);
      c[3] = __builtin_amdgcn_wmma_f32_16x16x4_f32(false, a, false, b3, (short)0, c[3], false, false);
    }
  }
}

// C/D layout: element r of v8f is row (lane>=16 ? 8 : 0)+r, col = lane&15.
#define TILE_STORE_IDX()                                        \
  const int lane = threadIdx.x & 31, wid = threadIdx.x >> 5;    \
  const int lr = (wid << 4) + ((lane >> 4) << 3);               \
  const int nbase = col0 + (lane & 15);

// ---------------- K1: per-head qkv projection --------------------------
__global__ __launch_bounds__(256) void k_proj(
    const float* __restrict__ q, const float* __restrict__ Wqkv,
    const float* __restrict__ bqkv, float* __restrict__ P)
{
  __shared__ float As[128 * AS_STRIDE];
  __shared__ float Bs[16 * BS_STRIDE];
  const int h = blockIdx.z;
  const int row0 = blockIdx.x << 7;
  const int col0 = blockIdx.y << 6;
  v8f c[4] = {{0,0,0,0,0,0,0,0},{0,0,0,0,0,0,0,0},
              {0,0,0,0,0,0,0,0},{0,0,0,0,0,0,0,0}};
  gemm_block_128x64(q, DIMS, Wqkv + (size_t)h * DIMS * PROJ_N, PROJ_N,
                    row0, col0, DIMS, As, Bs, c);
  TILE_STORE_IDX();
  const float* bq = bqkv + h * PROJ_N;
  float* Ph = P + (size_t)h * MROWS * PROJ_N;
  #pragma unroll
  for (int j = 0; j < 4; ++j) {
    const int n = nbase + 16 * j;
    #pragma unroll
    for (int r = 0; r < 8; ++r)
      Ph[(size_t)(row0 + lr + r) * PROJ_N + n] = c[j][r] + bq[n];
  }
}

// ---------------- K2: chunkwise retention scan -------------------------
// One WG per (b,h). 64x64 state distributed: thread (rg = tid>>6, c = tid&63)
// owns state[rg*16 .. rg*16+15][c]. Fuses GroupNorm sum/sumsq accumulation.
__global__ __launch_bounds__(256) void k_scan(
    const float* __restrict__ P, float* __restrict__ X,
    float* __restrict__ stats)
{
  const int b = blockIdx.x >> 3;
  const int h = blockIdx.x & 7;
  const float gamma  = 1.f - exp2f(-5.f - (float)h);
  const float l2g    = log2f(gamma);
  const float g2     = gamma * gamma;
  const float inv1mg = 1.f / (1.f - gamma);
  const int tid   = threadIdx.x;
  const int c     = tid & 63;
  const int rg    = tid >> 6;
  const int rbase = rg << 4;
  const int lane  = tid & 31, w = tid >> 5;
  const int pi = (w & 3) >> 1, pj = w & 1;            // q_i . k_j dot pairs
  const int pd = ((w >> 2) << 5) + lane;              // dim handled by lane

  __shared__ float rowbuf[2 * 192];   // chunk rows: [q(64) k(64) v(64)] x2
  __shared__ float red[2 * 4 * 64];   // cross-term partials / final reduce
  __shared__ float wsum[8];
  __shared__ float scn[4];

  const float* Ph = P + (size_t)h * MROWS * PROJ_N + (size_t)b * SEQ * PROJ_N;
  float st[16];
  #pragma unroll
  for (int r = 0; r < 16; ++r) st[r] = 0.f;

  float pf0 = Ph[tid];
  float pf1 = (tid < 128) ? Ph[256 + tid] : 0.f;
  float lsum = 0.f, lsq = 0.f;

  for (int t = 0; t < NCHUNK; ++t) {
    __syncthreads();                                  // prev reads of rowbuf done
    rowbuf[tid] = pf0;
    if (tid < 128) rowbuf[256 + tid] = pf1;
    __syncthreads();                                  // rowbuf visible
    if (t + 1 < NCHUNK) {                             // prefetch next chunk
      const float* nx = Ph + (size_t)(t + 1) * 384;
      pf0 = nx[tid];
      if (tid < 128) pf1 = nx[256 + tid];
    }
    const float* r0 = rowbuf;
    const float* r1 = rowbuf + 192;

    // cross term partials from OLD state: (q_i . state[:,c]) over my 16 rows
    float p0 = 0.f, p1 = 0.f;
    #pragma unroll
    for (int r = 0; r < 16; ++r) {
      const float s = st[r];
      p0 += r0[rbase + r] * s;
      p1 += r1[rbase + r] * s;
    }
    red[(0 * 4 + rg) * 64 + c] = p0;
    red[(1 * 4 + rg) * 64 + c] = p1;

    // 2x2 score dots: wave w -> pair (pi,pj), half-dim (w>>2); shfl reduce
    {
      const float* qi = (pi == 0) ? r0 : r1;
      const float* kj = (pj == 0) ? r0 : r1;
      float pr = qi[pd] * kj[64 + pd];
      #pragma unroll
      for (int m = 16; m > 0; m >>= 1) pr += __shfl_xor(pr, m, 32);
      if (lane == 0) wsum[w] = pr;
    }
    __syncthreads();
    if (tid == 0) {
      const float d00 = wsum[0] + wsum[4];
      const float d10 = wsum[2] + wsum[6];
      const float d11 = wsum[3] + wsum[7];
      // D slice at (t+1, t+1): D[a][bb] = gamma^(a-bb)[a>=bb] / sqrt(rowsum(bb))
      const float a1  = (float)(t + 2);               // exponent for col t+1
      const float rs0 = (1.f - exp2f(a1 * l2g)) * inv1mg;
      const float rs1 = (1.f - exp2f((a1 + 1.f) * l2g)) * inv1mg;
      const float i0 = rsqrtf(rs0), i1 = rsqrtf(rs1);
      const float isq = 0.125f;                       // 1/sqrt(HDIM)
      const float s00 = d00 * isq * i0;
      const float s10 = d10 * isq * gamma * i0;
      const float s11 = d11 * isq * i1;
      const float nr0 = fmaxf(fabsf(s00), 1.f);       // s01 == 0 (masked)
      const float nr1 = fmaxf(fabsf(s10 + s11), 1.f);
      scn[0] = s00 / nr0; scn[1] = 0.f;
      scn[2] = s10 / nr1; scn[3] = s11 / nr1;
    }
    __syncthreads();

    const float gpow = exp2f(-(float)t * l2g);        // gamma^(1-c), c=t+1
    const float gout = exp2f((float)(t + 2) * l2g);   // gamma^(c+1)
    const float v0c = r0[128 + c], v1c = r1[128 + c];
    if (rg < 2) {
      const float cross = red[(rg * 4 + 0) * 64 + c] + red[(rg * 4 + 1) * 64 + c]
                        + red[(rg * 4 + 2) * 64 + c] + red[(rg * 4 + 3) * 64 + c];
      const float o = scn[2 * rg] * v0c + scn[2 * rg + 1] * v1c + cross * gout;
      X[((size_t)b * SEQ + 2 * t + rg) * DIMS + h * HDIMS + c] = o;
      lsum += o; lsq += o * o;
    }
    // state update: st = st*g^2 + (k0^T v0 + k1^T v1) * gamma^(1-c)
    const float a0 = v0c * gpow, a1v = v1c * gpow;
    #pragma unroll
    for (int r = 0; r < 16; ++r) {
      const int rw = rbase + r;
      st[r] = st[r] * g2 + r0[64 + rw] * a0 + r1[64 + rw] * a1v;
    }
  }

  // GroupNorm stats for this (b,h): mean/var over SEQ*HDIM = 65536 elements
  __syncthreads();
  red[tid] = lsum;
  red[256 + tid] = lsq;
  __syncthreads();
  for (int off = 128; off > 0; off >>= 1) {
    if (tid < off) {
      red[tid] += red[tid + off];
      red[256 + tid] += red[256 + tid + off];
    }
    __syncthreads();
  }
  if (tid == 0) {
    const float inv = 1.f / 65536.f;
    const float mu  = red[0] * inv;
    const float var = red[256] * inv - mu * mu;
    stats[2 * ((b << 3) + h)]     = mu;
    stats[2 * ((b << 3) + h) + 1] = rsqrtf(var + 1e-3f);
  }
}

// ---------------- K3: swish gate * GroupNorm(X) ------------------------
__global__ __launch_bounds__(256) void k_gate(
    const float* __restrict__ q, const float* __restrict__ Wg,
    const float* __restrict__ bg, const float* __restrict__ X,
    const float* __restrict__ stats, const float* __restrict__ beta,
    float* __restrict__ Z)
{
  __shared__ float As[128 * AS_STRIDE];
  __shared__ float Bs[16 * BS_STRIDE];
  const int row0 = blockIdx.x << 7;
  const int col0 = blockIdx.y << 6;
  v8f c[4] = {{0,0,0,0,0,0,0,0},{0,0,0,0,0,0,0,0},
              {0,0,0,0,0,0,0,0},{0,0,0,0,0,0,0,0}};
  gemm_block_128x64(q, DIMS, Wg, DIMS, row0, col0, DIMS, As, Bs, c);
  TILE_STORE_IDX();
  const int hh = col0 >> 6;                 // whole 64-col tile is one head
  #pragma unroll
  for (int j = 0; j < 4; ++j) {
    const int n = nbase + 16 * j;
    const float bn = beta[n], bgn = bg[n];
    #pragma unroll
    for (int r = 0; r < 8; ++r) {
      const int m  = row0 + lr + r;
      const int bi = m >> 10;
      const float x = c[j][r] + bgn;
      const float g = x / (1.f + expf(-x));           // swish
      const float mu = stats[2 * ((bi << 3) + hh)];
      const float is = stats[2 * ((bi << 3) + hh) + 1];
      const float y = (X[(size_t)m * DIMS + n] - mu) * is + bn;
      Z[(size_t)m * DIMS + n] = g * y;
    }
  }
}

// ---------------- K4: output projection --------------------------------
__global__ __launch_bounds__(256) void k_outp(
    const float* __restrict__ Z, const float* __restrict__ Wo,
    const float* __restrict__ bo, float* __restrict__ out)
{
  __shared__ float As[128 * AS_STRIDE];
  __shared__ float Bs[16 * BS_STRIDE];
  const int row0 = blockIdx.x << 7;
  const int col0 = blockIdx.y << 6;
  v8f c[4] = {{0,0,0,0,0,0,0,0},{0,0,0,0,0,0,0,0},
              {0,0,0,0,0,0,0,0},{0,0,0,0,0,0,0,0}};
  gemm_block_128x64(Z, DIMS, Wo, DIMS, row0, col0, DIMS, As, Bs, c);
  TILE_STORE_IDX();
  #pragma unroll
  for (int j = 0; j < 4; ++j) {
    const int n = nbase + 16 * j;
    const float bon = bo[n];
    #pragma unroll
    for (int r = 0; r < 8; ++r)
      out[(size_t)(row0 + lr + r) * DIMS + n] = c[j][r] + bon;
  }
}

extern "C" void kernel_launch(void* const* d_in, const int* in_sizes, int n_in,
                              void* d_out, int out_size, void* d_ws, size_t ws_size,
                              hipStream_t stream) {
  (void)in_sizes; (void)n_in; (void)out_size; (void)ws_size;
  const float* q    = (const float*)d_in[0];
  // d_in[1], d_in[2] (k, v) are unused by the reference.
  const float* Wg   = (const float*)d_in[3];
  const float* bg   = (const float*)d_in[4];
  const float* Wo   = (const float*)d_in[5];
  const float* bo   = (const float*)d_in[6];
  const float* Wqkv = (const float*)d_in[7];
  const float* bqkv = (const float*)d_in[8];
  const float* beta = (const float*)d_in[9];
  float* out = (float*)d_out;

  float* P     = (float*)d_ws;                        // 25,165,824 f32
  float* X     = P + (size_t)NHEAD * MROWS * PROJ_N;  // 8,388,608 f32
  float* stats = X + (size_t)MROWS * DIMS;            // 256 f32
  float* Z     = P;                                   // alias: P dead after k_scan

  k_proj<<<dim3(MROWS / 128, PROJ_N / 64, NHEAD), 256, 0, stream>>>(q, Wqkv, bqkv, P);
  k_scan<<<dim3(BATCH * NHEAD), 256, 0, stream>>>(P, X, stats);
  k_gate<<<dim3(MROWS / 128, DIMS / 64), 256, 0, stream>>>(q, Wg, bg, X, stats, beta, Z);
  k_outp<<<dim3(MROWS / 128, DIMS / 64), 256, 0, stream>>>(Z, Wo, bo, out);
}